// SpectralConv_Nd_26250840113683
// MI455X (gfx1250) — compile-verified
//
#include <hip/hip_runtime.h>

// ---------------- CDNA5 WMMA types ----------------
typedef float v2f __attribute__((ext_vector_type(2)));
typedef float v8f __attribute__((ext_vector_type(8)));

// ---------------- problem constants ----------------
// B=16, C_IN=64, C_OUT=64, NX=NY=256, K0=K1=20
// kept x-frequencies: {0..19} U {236..255}  (40 total), y-frequencies {0..19}

// ---------------- workspace layout (float offsets) ----------------
#define OFF_TB1 0LL
#define SZ_TB1  (256LL*48)          // y-DFT matrix  [ny=256][48]  (20 re, 20 im, 8 pad)
#define OFF_TB2 (OFF_TB1+SZ_TB1)
#define SZ_TB2  (512LL*80)          // x-DFT matrix  [2*256][2*40] real-packed complex
#define OFF_TB4 (OFF_TB2+SZ_TB2)
#define SZ_TB4  (80LL*512)          // inverse-x synthesis [2*40][2*256]
#define OFF_TB5 (OFF_TB4+SZ_TB4)
#define SZ_TB5  (40LL*256)          // inverse-y synthesis [2*20][256] (c_ky & 1/65536 folded)
#define OFF_P   (OFF_TB5+SZ_TB5)    // = 104448 floats

#define SZ_C1 (262144LL*48)         // stage A out: [b,i,nx][48]
#define SZ_A2 (20480LL*512)         // stage B in:  [(b,i,ky)][Xr(256),Xi(256)]
#define SZ_C2 (20480LL*80)          // stage B out: [(b,i,ky)][Zr(40),Zi(40)]
#define SZ_B3 (800LL*128*128)       // per-mode weight matrices
#define SZ_F  (800LL*16*128)        // stage C out per mode [b][Fr(64),Fi(64)]
#define SZ_A3 SZ_F                  // stage C in  per mode [b][Zr_i(64),Zi_i(64)]
#define SZ_A4 (20480LL*80)          // stage D in:  [(b,o,ky)][Fr(40),Fi(40)]
#define SZ_C4 (20480LL*512)         // stage D out: [(b,o,ky)][Yr(256),Yi(256)]
#define SZ_A5 (262144LL*40)         // stage E in:  [(b,o,nx)][Yr(20),Yi(20)]

// region aliasing across stage lifetimes (C1/A2 die before B3/F/A4/A3; A5 reuses front)
#define OFF_C1 OFF_P
#define OFF_A2 (OFF_P + SZ_C1)
#define OFF_B3 OFF_P
#define OFF_F  (OFF_P + SZ_B3)
#define OFF_A4 (OFF_F + SZ_F)
#define OFF_A3 (OFF_A4 + SZ_A4)
#define OFF_C2 (OFF_P + SZ_C1 + SZ_A2)
#define OFF_C4 (OFF_C2 + SZ_C2)
#define OFF_A5 OFF_P
// total = (OFF_C4 + SZ_C4)*4 bytes ~= 141 MB

// exact phase: 2*pi*((a mod 256)/256) -- integer reduction keeps sinf/cosf accurate
__device__ __forceinline__ float angf(int a) {
  return 6.28318530717958647692f * (float)(a & 255) * (1.0f/256.0f);
}

// ---------------- twiddle / synthesis tables ----------------
__global__ void make_tables(float* __restrict__ ws) {
  const float S = 1.0f / 65536.0f;   // (1/sqrt(NX*NY))^2, ortho fwd+inv
  for (long long idx = (long long)blockIdx.x*blockDim.x + threadIdx.x;
       idx < OFF_P; idx += (long long)gridDim.x * blockDim.x) {
    float v = 0.0f;
    if (idx < OFF_TB2) {                       // TB1 [256][48]: Xr=sum x cos, Xi=-sum x sin
      long long t = idx - OFF_TB1;
      int ny = (int)(t / 48), c = (int)(t % 48);
      if      (c < 20) v =  cosf(angf(c * ny));
      else if (c < 40) v = -sinf(angf((c - 20) * ny));
      else             v = 0.0f;               // pad columns
    } else if (idx < OFF_TB4) {                // TB2 [512][80]: [[Tr,Ti],[-Ti,Tr]], T=e^{-i th}
      long long t = idx - OFF_TB2;
      int k = (int)(t / 80), c = (int)(t % 80);
      int nx = k & 255, j = c % 40;
      int kx = (j < 20) ? j : 216 + j;         // 236 + (j-20)
      float th = angf(kx * nx);
      float cs = cosf(th), sn = sinf(th);      // Tr=cs, Ti=-sn
      if (k < 256) v = (c < 40) ?  cs : -sn;
      else         v = (c < 40) ?  sn :  cs;
    } else if (idx < OFF_TB5) {                // TB4 [80][512]: [[Er,Ei],[-Ei,Er]], E=e^{+i th}
      long long t = idx - OFF_TB4;
      int k = (int)(t / 512), n = (int)(t % 512);
      int j = k % 40, nx = n & 255;
      int kx = (j < 20) ? j : 216 + j;
      float th = angf(kx * nx);
      float cs = cosf(th), sn = sinf(th);
      if (k < 40) v = (n < 256) ?  cs : sn;
      else        v = (n < 256) ? -sn : cs;
    } else {                                   // TB5 [40][256]: Hermitian inverse-y, scaled
      long long t = idx - OFF_TB5;
      int k = (int)(t / 256), ny = (int)(t % 256);
      int ky = k % 20;
      float c = (ky == 0) ? 1.0f : 2.0f;       // ky>0 modes mirrored => factor 2
      float th = angf(ky * ny);
      v = (k < 20) ? S * c * cosf(th) : -S * c * sinf(th);
    }
    ws[idx] = v;
  }
}

// ---------------- batched WMMA-f32 GEMM with LDS-staged B panels ----------------
// C[M,N] = A[M,K] * B[K,N], row-major. M%16==0, K%KB==0, KB%4==0.
// blockIdx.x : m-tile group (blockDim.x/32 waves, one 16-row tile each)
// blockIdx.y : N-group of width 16*NT (all waves in block share it -> shared B panel)
// blockIdx.z : batch
// B K-panels are staged into LDS COLUMN-MAJOR with +4-float padding:
//   smem[c*(KB+4) + r]  (c = column within group, r = k within panel)
// so each WMMA B fragment {B[ka][n], B[ka+1][n]} is one aligned ds_load_b64.
// Bank math: per-column stride KB+4 with gcd(KB+4,64)==4 (KB in {64,80,40}) puts
// lane (lg,li) on banks {(4*li + 2*lg + const) mod 64, +1} -> all 64 banks hit
// exactly once per fragment load, fully conflict-free, no register re-pairing.
// Fragment layouts per CDNA5 ISA 7.12.2 (wave32):
//   A 16x4 : lane l, vgpr v  -> A[m = l&15][k = v + 2*(l>>4)]
//   B 4x16 : lane l, vgpr v  -> B[k = v + 2*(l>>4)][n = l&15]
//   D 16x16: lane l, vgpr v  -> D[m = v + 8*(l>>4)][n = l&15]
template<int NT, int KB>
__global__ void gemm_wmma_f32_s(const float* __restrict__ A,
                                const float* __restrict__ B,
                                float* __restrict__ C,
                                int M, int N, int K,
                                long long sA, long long sB, long long sC) {
  constexpr int W   = 16 * NT;
  constexpr int LDK = KB + 4;                  // padded column stride (gcd(LDK,64)==4)
  __shared__ float smem[W * LDK];

  const float* Ab = A + (long long)blockIdx.z * sA;
  const float* Bb = B + (long long)blockIdx.z * sB;
  float*       Cb = C + (long long)blockIdx.z * sC;

  const int lane = threadIdx.x & 31;
  const int wpb  = blockDim.x >> 5;
  const int mt   = blockIdx.x * wpb + (threadIdx.x >> 5);
  const int colBase = blockIdx.y * W;
  const bool active = (mt < (M >> 4));         // wave-uniform
  const int row0 = mt << 4;
  const int lg = lane >> 4;
  const int li = lane & 15;

  v8f acc[NT];
#pragma unroll
  for (int t = 0; t < NT; ++t) {
    v8f z = {0.f,0.f,0.f,0.f,0.f,0.f,0.f,0.f};
    acc[t] = z;
  }

  const float* arow = Ab + (long long)(row0 + li) * K;   // only dereferenced if active

  for (int kp = 0; kp < K; kp += KB) {
    __syncthreads();
    for (int idx = threadIdx.x; idx < KB * W; idx += blockDim.x) {
      int r = idx / W, c = idx % W;            // coalesced global read over c
      smem[c * LDK + r] = Bb[(long long)(kp + r) * N + colBase + c];
    }
    __syncthreads();
    if (active) {                              // wave-uniform: EXEC all-ones inside
#pragma unroll 4
      for (int kk = 0; kk < KB; kk += 4) {
        const int ka = kk + (lg << 1);
        v2f a = *(const v2f*)&arow[kp + ka];   // aligned global b64 (K even, ka even)
#pragma unroll
        for (int t = 0; t < NT; ++t) {
          v2f b = *(const v2f*)&smem[(t * 16 + li) * LDK + ka];  // one ds_load_b64
          acc[t] = __builtin_amdgcn_wmma_f32_16x16x4_f32(
              false, a, false, b, (short)0, acc[t], false, false);
        }
      }
    }
  }

  if (active) {
#pragma unroll
    for (int t = 0; t < NT; ++t)
#pragma unroll
      for (int v = 0; v < 8; ++v)
        Cb[(long long)(row0 + v + (lg << 3)) * N + colBase + t * 16 + li] = acc[t][v];
  }
}

// ---------------- re-layout (gather) kernels ----------------
// A2[(b*64+i)*20+ky][nx | 256+nx] = C1[(b*64+i)*256+nx][ky | 20+ky]
__global__ void make_A2(const float* __restrict__ C1, float* __restrict__ A2) {
  long long n = 20480LL * 512;
  long long idx = (long long)blockIdx.x * blockDim.x + threadIdx.x;
  if (idx >= n) return;
  int r = (int)(idx / 512), col = (int)(idx % 512);
  int bi = r / 20, ky = r % 20;
  int nx = col & 255;
  int c  = (col < 256) ? ky : 20 + ky;
  A2[idx] = C1[((long long)bi * 256 + nx) * 48 + c];
}

// per-mode complex weight matrix: [[Wr,Wi],[-Wi,Wr]] (mode = j*20+ky, j<20 -> w0, else w1)
__global__ void make_B3(const float* __restrict__ w0re, const float* __restrict__ w0im,
                        const float* __restrict__ w1re, const float* __restrict__ w1im,
                        float* __restrict__ B3) {
  long long n = 800LL * 128 * 128;
  long long idx = (long long)blockIdx.x * blockDim.x + threadIdx.x;
  if (idx >= n) return;
  int mode = (int)(idx / 16384);
  int rem  = (int)(idx % 16384);
  int k = rem / 128, ncol = rem % 128;
  int j = mode / 20, ky = mode % 20;
  int i = k & 63, a = k >> 6;
  int o = ncol & 63, cb = ncol >> 6;
  const float* wre = (j < 20) ? w0re : w1re;
  const float* wim = (j < 20) ? w0im : w1im;
  int kxl = (j < 20) ? j : j - 20;
  long long widx = (((long long)i * 64 + o) * 20 + kxl) * 20 + ky;
  float Wr = wre[widx], Wi = wim[widx];
  B3[idx] = (a == 0) ? ((cb == 0) ? Wr : Wi)
                     : ((cb == 0) ? -Wi : Wr);
}

// A3[mode][b][i | 64+i] = C2[(b*64+i)*20+ky][j | 40+j]
__global__ void make_A3(const float* __restrict__ C2, float* __restrict__ A3) {
  long long n = 800LL * 16 * 128;
  long long idx = (long long)blockIdx.x * blockDim.x + threadIdx.x;
  if (idx >= n) return;
  int mode = (int)(idx / 2048);
  int rem  = (int)(idx % 2048);
  int b = rem / 128, k = rem % 128;
  int j = mode / 20, ky = mode % 20;
  int i = k & 63, a = k >> 6;
  A3[idx] = C2[((long long)(b * 64 + i) * 20 + ky) * 80 + ((a == 0) ? j : 40 + j)];
}

// A4[(b*64+o)*20+ky][j | 40+j] = F[(j*20+ky)*16+b][o | 64+o]
__global__ void make_A4(const float* __restrict__ F, float* __restrict__ A4) {
  long long n = 20480LL * 80;
  long long idx = (long long)blockIdx.x * blockDim.x + threadIdx.x;
  if (idx >= n) return;
  int r = (int)(idx / 80), c = (int)(idx % 80);
  int bo = r / 20, ky = r % 20;
  int j = c % 40, a = c / 40;
  int b = bo >> 6, o = bo & 63;
  A4[idx] = F[((long long)(j * 20 + ky) * 16 + b) * 128 + ((a == 0) ? o : 64 + o)];
}

// A5[(b*64+o)*256+nx][ky | 20+ky] = C4[(b*64+o)*20+ky][nx | 256+nx]
__global__ void make_A5(const float* __restrict__ C4, float* __restrict__ A5) {
  long long n = 262144LL * 40;
  long long idx = (long long)blockIdx.x * blockDim.x + threadIdx.x;
  if (idx >= n) return;
  int r = (int)(idx / 40), c = (int)(idx % 40);
  int bo = r >> 8, nx = r & 255;
  int ky = c % 20, a = c / 20;
  A5[idx] = C4[((long long)bo * 20 + ky) * 512 + ((a == 0) ? nx : 256 + nx)];
}

// ---------------- host orchestration ----------------
extern "C" void kernel_launch(void* const* d_in, const int* in_sizes, int n_in,
                              void* d_out, int out_size, void* d_ws, size_t ws_size,
                              hipStream_t stream) {
  (void)in_sizes; (void)n_in; (void)out_size; (void)ws_size;
  const float* x    = (const float*)d_in[0];
  const float* w0re = (const float*)d_in[1];
  const float* w0im = (const float*)d_in[2];
  const float* w1re = (const float*)d_in[3];
  const float* w1im = (const float*)d_in[4];
  float* ws  = (float*)d_ws;
  float* out = (float*)d_out;

  // twiddle + synthesis tables
  make_tables<<<128, 256, 0, stream>>>(ws);

  // Stage A: partial y-DFT  [262144 x 256] * [256 x 48]; NT=3 -> x streamed exactly once
  gemm_wmma_f32_s<3, 64><<<dim3(16384/8, 1, 1), 256, 0, stream>>>(
      x, ws + OFF_TB1, ws + OFF_C1, 262144, 48, 256, 0, 0, 0);

  make_A2<<<(int)((20480LL*512 + 255) / 256), 256, 0, stream>>>(ws + OFF_C1, ws + OFF_A2);

  // Stage B: partial x-DFT (real-packed complex)  [20480 x 512] * [512 x 80]; NT=5
  gemm_wmma_f32_s<5, 64><<<dim3(1280/8, 1, 1), 256, 0, stream>>>(
      ws + OFF_A2, ws + OFF_TB2, ws + OFF_C2, 20480, 80, 512, 0, 0, 0);

  make_B3<<<(int)((800LL*128*128 + 255) / 256), 256, 0, stream>>>(w0re, w0im, w1re, w1im, ws + OFF_B3);
  make_A3<<<(int)((800LL*16*128 + 255) / 256), 256, 0, stream>>>(ws + OFF_C2, ws + OFF_A3);

  // Stage C: per-mode complex channel mixing, 800 batched [16 x 128] * [128 x 128].
  // 256 threads: 1 wave computes (M=16 -> one tile), all 8 waves fill the LDS panel.
  gemm_wmma_f32_s<8, 64><<<dim3(1, 1, 800), 256, 0, stream>>>(
      ws + OFF_A3, ws + OFF_B3, ws + OFF_F, 16, 128, 128,
      2048LL, 16384LL, 2048LL);

  make_A4<<<(int)((20480LL*80 + 255) / 256), 256, 0, stream>>>(ws + OFF_F, ws + OFF_A4);

  // Stage D: inverse x synthesis  [20480 x 80] * [80 x 512]; NT=8, whole K in LDS
  gemm_wmma_f32_s<8, 80><<<dim3(1280/8, 4, 1), 256, 0, stream>>>(
      ws + OFF_A4, ws + OFF_TB4, ws + OFF_C4, 20480, 512, 80, 0, 0, 0);

  make_A5<<<(int)((262144LL*40 + 255) / 256), 256, 0, stream>>>(ws + OFF_C4, ws + OFF_A5);

  // Stage E: inverse y synthesis (Hermitian-weighted, normalized) -> d_out; NT=8
  gemm_wmma_f32_s<8, 40><<<dim3(16384/8, 2, 1), 256, 0, stream>>>(
      ws + OFF_A5, ws + OFF_TB5, out, 262144, 256, 40, 0, 0, 0);
}